// GraphSAGEEncoder_46866683134377
// MI455X (gfx1250) — compile-verified
//
#include <hip/hip_runtime.h>

#define N_NODES 100000
#define N_EDGES 600000
#define DIM 128
#define W_STRIDE 132   // padded LDS stride (floats): 4*n mod 64 -> conflict-free b64 B loads

typedef float v2f __attribute__((ext_vector_type(2)));
typedef float v8f __attribute__((ext_vector_type(8)));

// ---------------------------------------------------------------------------
// Utility: zero int buffer
// ---------------------------------------------------------------------------
__global__ void zero_ints_kernel(int* __restrict__ a, int n) {
    int i = blockIdx.x * blockDim.x + threadIdx.x;
    if (i < n) a[i] = 0;
}

// ---------------------------------------------------------------------------
// Histogram of destination nodes (int atomics only)
// ---------------------------------------------------------------------------
__global__ void hist_kernel(const int* __restrict__ dst, int* __restrict__ cnt, int n) {
    int i = blockIdx.x * blockDim.x + threadIdx.x;
    if (i < n) atomicAdd(&cnt[dst[i]], 1);
}

// ---------------------------------------------------------------------------
// Single-block exclusive scan (100001-entry row_ptr)
// ---------------------------------------------------------------------------
__global__ __launch_bounds__(1024) void scan_kernel(const int* __restrict__ cnt,
                                                    int* __restrict__ row_ptr, int n) {
    __shared__ int sdata[1024];
    __shared__ int s_carry;
    const int tid = threadIdx.x;
    if (tid == 0) s_carry = 0;
    __syncthreads();
    for (int base = 0; base < n; base += 1024) {
        int i = base + tid;
        int v = (i < n) ? cnt[i] : 0;
        sdata[tid] = v;
        __syncthreads();
        for (int off = 1; off < 1024; off <<= 1) {
            int t = (tid >= off) ? sdata[tid - off] : 0;
            __syncthreads();
            sdata[tid] += t;
            __syncthreads();
        }
        int incl = sdata[tid];
        int carry = s_carry;
        if (i < n) row_ptr[i] = carry + incl - v;
        int total = sdata[1023];
        __syncthreads();
        if (tid == 0) s_carry = carry + total;
        __syncthreads();
    }
    if (threadIdx.x == 0) row_ptr[n] = s_carry;
}

// ---------------------------------------------------------------------------
// Scatter edges into CSR-by-destination (col = src node)
// ---------------------------------------------------------------------------
__global__ void scatter_kernel(const int* __restrict__ src, const int* __restrict__ dst,
                               const int* __restrict__ row_ptr, int* __restrict__ fill,
                               int* __restrict__ csr, int n) {
    int i = blockIdx.x * blockDim.x + threadIdx.x;
    if (i < n) {
        int d = dst[i];
        int p = atomicAdd(&fill[d], 1);
        csr[row_ptr[d] + p] = src[i];
    }
}

// ---------------------------------------------------------------------------
// Mean aggregation: one wave32 per destination node; each lane owns 4 of the
// 128 features (float4 -> 512B coalesced per neighbor row, L2-resident).
// 4-way unrolled for memory-level parallelism; no float atomics.
// ---------------------------------------------------------------------------
__global__ __launch_bounds__(256) void aggregate_mean_kernel(
    const float* __restrict__ H, const int* __restrict__ row_ptr,
    const int* __restrict__ csr, float* __restrict__ meanout, int n_nodes) {
    int gwave = (blockIdx.x * blockDim.x + threadIdx.x) >> 5;
    int lane = threadIdx.x & 31;
    if (gwave >= n_nodes) return;
    int beg = row_ptr[gwave];
    int end = row_ptr[gwave + 1];
    float4 acc = make_float4(0.f, 0.f, 0.f, 0.f);
    const float* Hl = H + lane * 4;
    int e = beg;
    for (; e + 3 < end; e += 4) {
        int s0 = csr[e + 0];
        int s1 = csr[e + 1];
        int s2 = csr[e + 2];
        int s3 = csr[e + 3];
        float4 v0 = *(const float4*)(Hl + (size_t)s0 * DIM);
        float4 v1 = *(const float4*)(Hl + (size_t)s1 * DIM);
        float4 v2 = *(const float4*)(Hl + (size_t)s2 * DIM);
        float4 v3 = *(const float4*)(Hl + (size_t)s3 * DIM);
        acc.x += (v0.x + v1.x) + (v2.x + v3.x);
        acc.y += (v0.y + v1.y) + (v2.y + v3.y);
        acc.z += (v0.z + v1.z) + (v2.z + v3.z);
        acc.w += (v0.w + v1.w) + (v2.w + v3.w);
    }
    for (; e < end; ++e) {
        int s = csr[e];
        float4 v = *(const float4*)(Hl + (size_t)s * DIM);
        acc.x += v.x; acc.y += v.y; acc.z += v.z; acc.w += v.w;
    }
    int deg = end - beg;
    float inv = 1.0f / (float)(deg > 0 ? deg : 1);
    acc.x *= inv; acc.y *= inv; acc.z *= inv; acc.w *= inv;
    *(float4*)(meanout + (size_t)gwave * DIM + lane * 4) = acc;
}

// ---------------------------------------------------------------------------
// Fused SAGE linear: out = [relu]( Amean @ Wl + bl + Aself @ Wr )
// 8 waves/block, each wave -> 16x128 output strip via V_WMMA_F32_16X16X4_F32.
// Two phases reuse one 128x132-float LDS tile (Wl then Wr, transposed).
// A fragment: one unconditional clamped global_load_b64 per k-step (L2 hit);
// B fragments batched 8-at-a-time in registers so DS loads pipeline ahead of
// the WMMAs. No local arrays with runtime indices -> nothing spills to scratch.
// ---------------------------------------------------------------------------
__global__ __launch_bounds__(256) void sage_gemm_wmma_kernel(
    const float* __restrict__ Amean, const float* __restrict__ Aself,
    const float* __restrict__ Wl, const float* __restrict__ bl,
    const float* __restrict__ Wr, float* __restrict__ out,
    int n_rows, int do_relu) {
    extern __shared__ float smem[];          // [128][W_STRIDE] Wt + [128] bias
    float* Wt = smem;                        // Wt[n*W_STRIDE + k] = W[k*128 + n]
    float* bias = smem + DIM * W_STRIDE;

    const int tid  = threadIdx.x;
    const int wave = tid >> 5;               // 0..7
    const int lane = tid & 31;
    const int half = lane >> 4;              // 0/1 (K sub-pair select)
    const int lr   = lane & 15;              // row-in-tile (A/D) == col-in-tile (B/D)

    const int m0 = blockIdx.x * 128 + wave * 16;

    if (tid < DIM) bias[tid] = bl[tid];

    v8f acc[8];
#pragma unroll
    for (int t = 0; t < 8; ++t)
#pragma unroll
        for (int j = 0; j < 8; ++j) acc[t][j] = 0.0f;

    // Clamp: an OOB A-row only corrupts a D-row the epilogue never stores,
    // so no predication is needed -> plain global_load_b64 for A fragments.
    int rowc = m0 + lr;
    if (rowc > n_rows - 1) rowc = n_rows - 1;

    for (int pass = 0; pass < 2; ++pass) {
        const float* W = pass ? Wr : Wl;
        const float* A = pass ? Aself : Amean;

        __syncthreads();                     // previous phase's Wt reads done
        // Stage W transposed into LDS: b128 coalesced reads, b32 scatter.
        for (int idx = tid * 4; idx < DIM * DIM; idx += 256 * 4) {
            float4 w = *(const float4*)(W + idx);
            int k = idx >> 7;                // row of W
            int n = idx & 127;               // first of 4 consecutive cols
            Wt[(n + 0) * W_STRIDE + k] = w.x;
            Wt[(n + 1) * W_STRIDE + k] = w.y;
            Wt[(n + 2) * W_STRIDE + k] = w.z;
            Wt[(n + 3) * W_STRIDE + k] = w.w;
        }
        __syncthreads();

        if (m0 < n_rows) {                   // wave-uniform: EXEC all-1 inside
            const float* Arow = A + (size_t)rowc * DIM + 2 * half;
            const float* Wh = Wt + lr * W_STRIDE + 2 * half;
#pragma unroll 4
            for (int ks = 0; ks < 32; ++ks) {
                const int k0 = ks * 4;
                // A 16x4 f32 fragment: v0 = K(k0+2h), v1 = K(k0+2h+1)
                v2f a = *(const v2f*)(Arow + k0);
                v2f b[8];
#pragma unroll
                for (int t = 0; t < 8; ++t)
                    b[t] = *(const v2f*)(Wh + t * 16 * W_STRIDE + k0);
#pragma unroll
                for (int t = 0; t < 8; ++t)
                    acc[t] = __builtin_amdgcn_wmma_f32_16x16x4_f32(
                        /*neg_a=*/false, a, /*neg_b=*/false, b[t],
                        /*c_mod=*/(short)0, acc[t],
                        /*reuse_a=*/false, /*reuse_b=*/false);
            }
        }
    }

    // Epilogue: D 16x16 f32 layout: vgpr j -> M = j + 8*half, N = lr (per tile)
    if (m0 < n_rows) {
#pragma unroll
        for (int t = 0; t < 8; ++t) {
            int col = t * 16 + lr;
            float bv = bias[col];
#pragma unroll
            for (int j = 0; j < 8; ++j) {
                int row = m0 + half * 8 + j;
                if (row < n_rows) {
                    float v = acc[t][j] + bv;
                    if (do_relu) v = fmaxf(v, 0.0f);
                    out[(size_t)row * DIM + col] = v;
                }
            }
        }
    }
}

// ---------------------------------------------------------------------------
// Host-side orchestration (graph-capture safe: stream-only launches)
// ---------------------------------------------------------------------------
extern "C" void kernel_launch(void* const* d_in, const int* in_sizes, int n_in,
                              void* d_out, int out_size, void* d_ws, size_t ws_size,
                              hipStream_t stream) {
    const float* x    = (const float*)d_in[0];
    const int*   eidx = (const int*)d_in[1];
    const int*   src  = eidx;
    const int*   dst  = eidx + N_EDGES;
    const float* Wl1  = (const float*)d_in[2];
    const float* bl1  = (const float*)d_in[3];
    const float* Wr1  = (const float*)d_in[4];
    const float* Wl2  = (const float*)d_in[5];
    const float* bl2  = (const float*)d_in[6];
    const float* Wr2  = (const float*)d_in[7];
    float* out = (float*)d_out;

    // Workspace carve-up (all regions fully written before read)
    float* meanbuf = (float*)d_ws;                       // N_NODES*DIM f32
    float* h1      = meanbuf + (size_t)N_NODES * DIM;    // N_NODES*DIM f32
    int*   cnt     = (int*)(h1 + (size_t)N_NODES * DIM); // N_NODES
    int*   fill    = cnt + N_NODES;                      // N_NODES (contiguous w/ cnt)
    int*   row_ptr = fill + N_NODES;                     // N_NODES+1
    int*   csr     = row_ptr + N_NODES + 4;              // N_EDGES (aligned)

    const int TB = 256;
    dim3 blkE((N_EDGES + TB - 1) / TB);
    dim3 blkZ((2 * N_NODES + TB - 1) / TB);
    dim3 blkAgg((N_NODES * 32 + TB - 1) / TB);           // one wave32 per node
    dim3 blkGemm((N_NODES + 127) / 128);
    size_t gemm_lds = (size_t)(DIM * W_STRIDE + DIM) * sizeof(float); // ~66.5 KB

    // Build CSR by destination (reused by both layers)
    zero_ints_kernel<<<blkZ, TB, 0, stream>>>(cnt, 2 * N_NODES);     // cnt + fill
    hist_kernel<<<blkE, TB, 0, stream>>>(dst, cnt, N_EDGES);
    scan_kernel<<<1, 1024, 0, stream>>>(cnt, row_ptr, N_NODES);
    scatter_kernel<<<blkE, TB, 0, stream>>>(src, dst, row_ptr, fill, csr, N_EDGES);

    // Layer 1: mean-agg(x) -> GEMM -> relu -> h1
    aggregate_mean_kernel<<<blkAgg, TB, 0, stream>>>(x, row_ptr, csr, meanbuf, N_NODES);
    sage_gemm_wmma_kernel<<<blkGemm, TB, gemm_lds, stream>>>(
        meanbuf, x, Wl1, bl1, Wr1, h1, N_NODES, /*relu=*/1);

    // Layer 2: mean-agg(h1) -> GEMM -> out
    aggregate_mean_kernel<<<blkAgg, TB, 0, stream>>>(h1, row_ptr, csr, meanbuf, N_NODES);
    sage_gemm_wmma_kernel<<<blkGemm, TB, gemm_lds, stream>>>(
        meanbuf, h1, Wl2, bl2, Wr2, out, N_NODES, /*relu=*/0);
}